// FFJORD_6528350290458
// MI455X (gfx1250) — compile-verified
//
#include <hip/hip_runtime.h>
#include <hip/hip_bf16.h>

// ---------------------------------------------------------------------------
// FFJORD on gfx1250: persistent per-batch-tile kernel, all 6 GEMMs per RK4
// stage done with V_WMMA_F32_16X16X4_F32 (exact fp32 on the matrix pipe).
// ---------------------------------------------------------------------------

#define BDIM    4096
#define DDIM    64
#define CDIM    16
#define HDIM    512
#define NBIJ    2
#define NSTEPS  8
#define MT      32        // batch rows per workgroup
#define NTHREADS 256      // 8 waves (wave32)

// LDS strides (floats) -- even so float2 fragment loads stay 8B-aligned
#define SH 516            // [MT][HDIM] buffers
#define SZ 84             // z = [y|cond|t|pad], K padded 81 -> 84
#define SY 68             // [MT][DDIM] buffers

// prepped-weight workspace layout (floats, per bijector)
#define OFF_W1P  0
#define SZ_W1P   (84*512)
#define OFF_W3T  (OFF_W1P + SZ_W1P)
#define SZ_W3T   (64*512)
#define OFF_W2T  (OFF_W3T + SZ_W3T)
#define SZ_W2T   (512*512)
#define OFF_W1YT (OFF_W2T + SZ_W2T)
#define SZ_W1YT  (512*64)
#define WS_PER   (OFF_W1YT + SZ_W1YT)

#define SMEM_FLOATS (2*MT*SH + MT*SZ + 6*MT*SY + MT*CDIM + 2*MT)

typedef float v2f __attribute__((ext_vector_type(2)));
typedef float v8f __attribute__((ext_vector_type(8)));

#define V8ZERO {0.f,0.f,0.f,0.f,0.f,0.f,0.f,0.f}

__device__ __forceinline__ v8f wmma4(v2f a, v2f b, v8f c) {
  // D = A(16x4,f32) x B(4x16,f32) + C(16x16,f32)
  return __builtin_amdgcn_wmma_f32_16x16x4_f32(false, a, false, b,
                                               (short)0, c, false, false);
}

// C = A[32 x K](LDS) * B[K x Bn](global, row-major) tile at columns n0..n0+15.
// Two 16-row fragments share each B fragment (2x reuse of the weight stream).
__device__ __forceinline__ void gemm_m32(const float* A, int As,
                                         const float* __restrict__ B, int Bn,
                                         int K, int n0, int lmod, int lhalf,
                                         v8f& d0, v8f& d1) {
  const float* arow0 = A + lmod * As;
  const float* arow1 = A + (16 + lmod) * As;
  const float* bcol  = B + n0 + lmod;
#pragma unroll 4
  for (int k = 0; k < K; k += 4) {
    const int ka = k + 2 * lhalf;
    v2f a0 = *(const v2f*)(arow0 + ka);
    v2f a1 = *(const v2f*)(arow1 + ka);
    v2f b;
    b.x = bcol[(size_t)ka * Bn];
    b.y = bcol[(size_t)(ka + 1) * Bn];
    d0 = wmma4(a0, b, d0);
    d1 = wmma4(a1, b, d1);
  }
}

// Single 16-row fragment variant (used for the N=64 GEMMs).
__device__ __forceinline__ void gemm_m16(const float* A, int As,
                                         const float* __restrict__ B, int Bn,
                                         int K, int n0, int rb, int lmod, int lhalf,
                                         v8f& d) {
  const float* arow = A + (rb + lmod) * As;
  const float* bcol = B + n0 + lmod;
#pragma unroll 4
  for (int k = 0; k < K; k += 4) {
    const int ka = k + 2 * lhalf;
    v2f a = *(const v2f*)(arow + ka);
    v2f b;
    b.x = bcol[(size_t)ka * Bn];
    b.y = bcol[(size_t)(ka + 1) * Bn];
    d = wmma4(a, b, d);
  }
}

// ---------------------------------------------------------------------------
// Prep: build padded W1 (K 81->84, zero rows), W3^T, W2^T, W1[0:64,:]^T in ws.
// ---------------------------------------------------------------------------
__global__ void ffjord_prep_kernel(const float* __restrict__ W1,
                                   const float* __restrict__ W2,
                                   const float* __restrict__ W3,
                                   float* __restrict__ ws) {
  const size_t total = (size_t)NBIJ * WS_PER;
  for (size_t idx = (size_t)blockIdx.x * blockDim.x + threadIdx.x; idx < total;
       idx += (size_t)gridDim.x * blockDim.x) {
    const int ib = (int)(idx / WS_PER);
    const int r  = (int)(idx % WS_PER);
    float v;
    if (r < OFF_W3T) {                       // W1p [84][512]
      const int k = r >> 9, n = r & 511;
      v = (k < 81) ? W1[(size_t)ib * 81 * 512 + (size_t)k * 512 + n] : 0.f;
    } else if (r < OFF_W2T) {                // W3T [64][512] = W3^T
      const int q = r - OFF_W3T;
      const int d = q >> 9, h = q & 511;
      v = W3[(size_t)ib * 512 * 64 + (size_t)h * 64 + d];
    } else if (r < OFF_W1YT) {               // W2T [512][512] = W2^T
      const int q = r - OFF_W2T;
      const int a = q >> 9, b = q & 511;
      v = W2[(size_t)ib * 512 * 512 + (size_t)b * 512 + a];
    } else {                                 // W1yT [512][64] = W1[0:64,:]^T
      const int q = r - OFF_W1YT;
      const int h = q >> 6, d = q & 63;
      v = W1[(size_t)ib * 81 * 512 + (size_t)d * 512 + h];
    }
    ws[idx] = v;
  }
}

// ---------------------------------------------------------------------------
// Main persistent kernel: one workgroup integrates 32 batch rows end to end.
// ---------------------------------------------------------------------------
__global__ __launch_bounds__(NTHREADS)
void ffjord_kernel(const float* __restrict__ x, const float* __restrict__ cond,
                   const float* __restrict__ eps,
                   const float* __restrict__ b1, const float* __restrict__ b2,
                   const float* __restrict__ b3,
                   const float* __restrict__ W2all, const float* __restrict__ W3all,
                   const float* __restrict__ ws, float* __restrict__ out) {
  extern __shared__ float smem[];
  float* h1 = smem;                  // [MT][SH]  h1 / later g1
  float* h2 = h1 + MT * SH;          // [MT][SH]  h2 / later g2
  float* zb = h2 + MT * SH;          // [MT][SZ]  [y|cond|t|0]
  float* yb = zb + MT * SZ;          // [MT][SY]  y
  float* yt = yb + MT * SY;          // [MT][SY]  stage input
  float* af = yt + MT * SY;          // [MT][SY]  RK4 f accumulator
  float* fb = af + MT * SY;          // [MT][SY]  f = MLP output
  float* gj = fb + MT * SY;          // [MT][SY]  eps^T J
  float* ep = gj + MT * SY;          // [MT][SY]  Hutchinson probe
  float* cd = ep + MT * SY;          // [MT][CDIM]
  float* al = cd + MT * CDIM;        // [MT] logdet stage accumulator
  float* lv = al + MT;               // [MT] logdet

  const int tid   = threadIdx.x;
  const int wave  = tid >> 5;
  const int lane  = tid & 31;
  const int lmod  = lane & 15;
  const int lhalf = lane >> 4;
  const int b0    = blockIdx.x * MT;

  const float dt  = 1.0f / NSTEPS;
  const float dt6 = dt / 6.0f;

  for (int i = tid; i < MT * DDIM; i += NTHREADS) {
    const int r = i >> 6, c = i & 63;
    yb[r * SY + c] = x[(size_t)(b0 + r) * DDIM + c];
  }
  for (int i = tid; i < MT * CDIM; i += NTHREADS) {
    const int r = i >> 4, c = i & 15;
    cd[i] = cond[(size_t)(b0 + r) * CDIM + c];
  }
  if (tid < MT) lv[tid] = 0.f;

  for (int ib = 0; ib < NBIJ; ++ib) {
    const float* wsb  = ws + (size_t)ib * WS_PER;
    const float* W1p  = wsb + OFF_W1P;
    const float* W3T  = wsb + OFF_W3T;
    const float* W2T  = wsb + OFF_W2T;
    const float* W1yT = wsb + OFF_W1YT;
    const float* W2g  = W2all + (size_t)ib * 512 * 512;
    const float* W3g  = W3all + (size_t)ib * 512 * 64;
    const float* b1g  = b1 + ib * 512;
    const float* b2g  = b2 + ib * 512;
    const float* b3g  = b3 + ib * 64;

    for (int i = tid; i < MT * DDIM; i += NTHREADS) {
      const int r = i >> 6, c = i & 63;
      ep[r * SY + c] = eps[((size_t)ib * BDIM + (b0 + r)) * DDIM + c];
    }
    __syncthreads();

    for (int step = 0; step < NSTEPS; ++step) {
      const float t0 = step * dt;
      for (int i = tid; i < MT * DDIM; i += NTHREADS) {
        const int r = i >> 6, c = i & 63;
        af[r * SY + c] = 0.f;
      }
      if (tid < MT) al[tid] = 0.f;
      __syncthreads();

      for (int s = 0; s < 4; ++s) {
        const float ts = t0 + ((s == 0) ? 0.f : (s == 3) ? dt : 0.5f * dt);
        const float wf = (s == 0 || s == 3) ? 1.f : 2.f;
        const float* yin = (s == 0) ? yb : yt;

        // z = [y | cond | t | 0]
        for (int i = tid; i < MT * SZ; i += NTHREADS) {
          const int r = i / SZ, c = i % SZ;
          float v;
          if (c < DDIM)             v = yin[r * SY + c];
          else if (c < DDIM + CDIM) v = cd[r * CDIM + (c - DDIM)];
          else if (c == DDIM + CDIM) v = ts;
          else                      v = 0.f;
          zb[i] = v;
        }
        __syncthreads();

        // layer 1: h1 = tanh(z @ W1p + b1)           [32x512], K=84
        for (int nt = wave * 4; nt < wave * 4 + 4; ++nt) {
          const int n0 = nt * 16, col = n0 + lmod;
          v8f d0 = V8ZERO, d1 = V8ZERO;
          gemm_m32(zb, SZ, W1p, HDIM, 84, n0, lmod, lhalf, d0, d1);
          const float bias = b1g[col];
#pragma unroll
          for (int v = 0; v < 8; ++v) {
            const int r = 8 * lhalf + v;
            h1[r * SH + col]        = tanhf(d0[v] + bias);
            h1[(r + 16) * SH + col] = tanhf(d1[v] + bias);
          }
        }
        __syncthreads();

        // layer 2: h2 = tanh(h1 @ W2 + b2)           [32x512], K=512
        for (int nt = wave * 4; nt < wave * 4 + 4; ++nt) {
          const int n0 = nt * 16, col = n0 + lmod;
          v8f d0 = V8ZERO, d1 = V8ZERO;
          gemm_m32(h1, SH, W2g, HDIM, HDIM, n0, lmod, lhalf, d0, d1);
          const float bias = b2g[col];
#pragma unroll
          for (int v = 0; v < 8; ++v) {
            const int r = 8 * lhalf + v;
            h2[r * SH + col]        = tanhf(d0[v] + bias);
            h2[(r + 16) * SH + col] = tanhf(d1[v] + bias);
          }
        }
        __syncthreads();

        // layer 3: f = h2 @ W3 + b3                  [32x64], K=512
        {
          const int n0 = (wave & 3) * 16, rb = (wave >> 2) * 16, col = n0 + lmod;
          v8f d = V8ZERO;
          gemm_m16(h2, SH, W3g, DDIM, HDIM, n0, rb, lmod, lhalf, d);
          const float bias = b3g[col];
#pragma unroll
          for (int v = 0; v < 8; ++v)
            fb[(rb + 8 * lhalf + v) * SY + col] = d[v] + bias;
        }
        __syncthreads();

        // vjp 1: g2 = (eps @ W3^T) * (1 - h2^2)      [32x512], K=64 (in place)
        for (int nt = wave * 4; nt < wave * 4 + 4; ++nt) {
          const int n0 = nt * 16, col = n0 + lmod;
          v8f d0 = V8ZERO, d1 = V8ZERO;
          gemm_m32(ep, SY, W3T, HDIM, DDIM, n0, lmod, lhalf, d0, d1);
#pragma unroll
          for (int v = 0; v < 8; ++v) {
            const int r = 8 * lhalf + v;
            const float a = h2[r * SH + col];
            h2[r * SH + col] = d0[v] * (1.f - a * a);
            const float b = h2[(r + 16) * SH + col];
            h2[(r + 16) * SH + col] = d1[v] * (1.f - b * b);
          }
        }
        __syncthreads();

        // vjp 2: g1 = (g2 @ W2^T) * (1 - h1^2)       [32x512], K=512 (in place)
        for (int nt = wave * 4; nt < wave * 4 + 4; ++nt) {
          const int n0 = nt * 16, col = n0 + lmod;
          v8f d0 = V8ZERO, d1 = V8ZERO;
          gemm_m32(h2, SH, W2T, HDIM, HDIM, n0, lmod, lhalf, d0, d1);
#pragma unroll
          for (int v = 0; v < 8; ++v) {
            const int r = 8 * lhalf + v;
            const float a = h1[r * SH + col];
            h1[r * SH + col] = d0[v] * (1.f - a * a);
            const float b = h1[(r + 16) * SH + col];
            h1[(r + 16) * SH + col] = d1[v] * (1.f - b * b);
          }
        }
        __syncthreads();

        // vjp 3: epsJ = g1 @ W1[0:64,:]^T            [32x64], K=512
        {
          const int n0 = (wave & 3) * 16, rb = (wave >> 2) * 16, col = n0 + lmod;
          v8f d = V8ZERO;
          gemm_m16(h1, SH, W1yT, DDIM, HDIM, n0, rb, lmod, lhalf, d);
#pragma unroll
          for (int v = 0; v < 8; ++v)
            gj[(rb + 8 * lhalf + v) * SY + col] = d[v];
        }
        __syncthreads();

        // RK4 stage bookkeeping
        for (int i = tid; i < MT * DDIM; i += NTHREADS) {
          const int r = i >> 6, c = i & 63;
          const float f = fb[r * SY + c];
          af[r * SY + c] += wf * f;
          if (s < 3) {
            const float cn = (s == 2) ? dt : 0.5f * dt;
            yt[r * SY + c] = yb[r * SY + c] + cn * f;
          }
        }
        if (tid < MT) {
          float sum = 0.f;
#pragma unroll 8
          for (int c = 0; c < DDIM; ++c)
            sum += gj[tid * SY + c] * ep[tid * SY + c];
          al[tid] += wf * sum;
        }
        __syncthreads();
      } // stages

      for (int i = tid; i < MT * DDIM; i += NTHREADS) {
        const int r = i >> 6, c = i & 63;
        yb[r * SY + c] += dt6 * af[r * SY + c];
      }
      if (tid < MT) lv[tid] += dt6 * al[tid];
      __syncthreads();
    } // steps
  } // bijectors

  // out[B, 65] = [y | logdet]
  for (int i = tid; i < MT * (DDIM + 1); i += NTHREADS) {
    const int r = i / (DDIM + 1), c = i % (DDIM + 1);
    out[(size_t)(b0 + r) * (DDIM + 1) + c] = (c < DDIM) ? yb[r * SY + c] : lv[r];
  }
}

extern "C" void kernel_launch(void* const* d_in, const int* in_sizes, int n_in,
                              void* d_out, int out_size, void* d_ws, size_t ws_size,
                              hipStream_t stream) {
  const float* x    = (const float*)d_in[0];
  const float* cond = (const float*)d_in[1];
  const float* eps  = (const float*)d_in[2];
  const float* W1   = (const float*)d_in[3];
  const float* b1   = (const float*)d_in[4];
  const float* W2   = (const float*)d_in[5];
  const float* b2   = (const float*)d_in[6];
  const float* W3   = (const float*)d_in[7];
  const float* b3   = (const float*)d_in[8];
  float* out = (float*)d_out;
  float* ws  = (float*)d_ws;

  ffjord_prep_kernel<<<512, 256, 0, stream>>>(W1, W2, W3, ws);

  const size_t shbytes = (size_t)SMEM_FLOATS * sizeof(float); // ~197 KB < 320 KB/WG
  ffjord_kernel<<<BDIM / MT, NTHREADS, shbytes, stream>>>(
      x, cond, eps, b1, b2, b3, W2, W3, ws, out);
}